// VisionTextMatching_28037546508716
// MI455X (gfx1250) — compile-verified
//
#include <hip/hip_runtime.h>
#include <hip/hip_bf16.h>
#include <math.h>

#define EPS 1e-8f

typedef __attribute__((ext_vector_type(2))) float v2f;
typedef __attribute__((ext_vector_type(8))) float v8f;

// ---------------------------------------------------------------------------
// Kernel 1: text projection + bias + row L2-normalize.
//   tn[m][d] = normalize_row( text_pooled[m][:768] . proj_w[d][:768] + b[d] )
// One block of 128 threads per text row m; text row staged in LDS.
// ---------------------------------------------------------------------------
__global__ __launch_bounds__(128)
void proj_norm_kernel(const float* __restrict__ text_pooled,
                      const float* __restrict__ proj_w,
                      const float* __restrict__ proj_b,
                      float* __restrict__ tn) {
    __shared__ __align__(16) float trow[768];
    __shared__ float wsum[4];
    const int m   = blockIdx.x;
    const int tid = threadIdx.x;

    // Stage text_pooled row into LDS (768 floats, 128 threads -> 6 each).
    const float4* src4 = (const float4*)(text_pooled + (size_t)m * 768);
    float4* trow4 = (float4*)trow;
    for (int i = tid; i < 192; i += 128) trow4[i] = src4[i];
    __syncthreads();

    // Each thread computes one output feature d = tid (D = 128).
    const float4* w4 = (const float4*)(proj_w + (size_t)tid * 768);
    float acc = 0.0f;
    #pragma unroll 4
    for (int k = 0; k < 192; ++k) {
        float4 a = trow4[k];
        float4 b = w4[k];
        acc += a.x * b.x + a.y * b.y + a.z * b.z + a.w * b.w;
    }
    acc += proj_b[tid];

    // Row-wise sum of squares across the 128 threads (4 waves of 32).
    float ss = acc * acc;
    #pragma unroll
    for (int off = 16; off > 0; off >>= 1) ss += __shfl_xor(ss, off, 32);
    if ((tid & 31) == 0) wsum[tid >> 5] = ss;
    __syncthreads();
    const float tot = wsum[0] + wsum[1] + wsum[2] + wsum[3];
    const float inv = 1.0f / fmaxf(sqrtf(tot), EPS);

    tn[(size_t)m * 128 + tid] = acc * inv;
}

// ---------------------------------------------------------------------------
// Kernel 2: vision row L2-normalize. One wave (32 lanes) per row, float4/lane.
// ---------------------------------------------------------------------------
__global__ __launch_bounds__(256)
void vision_norm_kernel(const float* __restrict__ vf,
                        float* __restrict__ vn) {
    const int tid  = threadIdx.x;
    const int lane = tid & 31;
    const int row  = blockIdx.x * 8 + (tid >> 5);

    const float4* src = (const float4*)(vf + (size_t)row * 128);
    float4 v = src[lane];
    float ss = v.x * v.x + v.y * v.y + v.z * v.z + v.w * v.w;
    #pragma unroll
    for (int off = 16; off > 0; off >>= 1) ss += __shfl_xor(ss, off, 32);
    const float inv = 1.0f / fmaxf(sqrtf(ss), EPS);
    float4 o = make_float4(v.x * inv, v.y * inv, v.z * inv, v.w * inv);
    ((float4*)(vn + (size_t)row * 128))[lane] = o;
}

// ---------------------------------------------------------------------------
// Kernel 3: sim = vn @ tn^T  (8192 x 8192, K = 128) via V_WMMA_F32_16X16X4_F32.
// Block: 256 threads (8 waves), 128x128 output tile, K chunked by 32 via LDS.
// Wave layout: 4x2 grid of waves, each wave owns 32 rows x 64 cols = 2x4
// WMMA 16x16 tiles.
// ---------------------------------------------------------------------------
#define KC    32
#define LDSW  36   // 32 + 4 pad: (36*m mod 64) is a permutation -> no bank conflicts

__global__ __launch_bounds__(256)
void cosine_gemm_kernel(const float* __restrict__ vn,
                        const float* __restrict__ tn,
                        float* __restrict__ out) {
    __shared__ __align__(16) float As[128 * LDSW];
    __shared__ __align__(16) float Bs[128 * LDSW];

    const int tid  = threadIdx.x;
    const int lane = tid & 31;
    const int w    = tid >> 5;          // wave 0..7
    const int wr   = w >> 1;            // wave row 0..3  (32 rows each)
    const int wc   = w & 1;             // wave col 0..1  (64 cols each)

    const int rowBase = blockIdx.y * 128;   // vision rows
    const int colBase = blockIdx.x * 128;   // text rows (output cols)

    const int mfrag = lane & 15;        // M (or N) index within 16
    const int hi    = lane >> 4;        // selects K sub-pair {0,1} vs {2,3}

    v8f acc[2][4];
    #pragma unroll
    for (int i = 0; i < 2; ++i)
        #pragma unroll
        for (int j = 0; j < 4; ++j)
            acc[i][j] = (v8f)(0.0f);

    for (int kc = 0; kc < 128; kc += KC) {
        __syncthreads();
        // Cooperative load: 128 rows x 32 K-floats for A and B = 1024 float4
        // each; 256 threads -> 4 float4 per matrix per thread.
        #pragma unroll
        for (int i = 0; i < 4; ++i) {
            const int q  = tid + i * 256;      // 0..1023
            const int r  = q >> 3;             // 0..127 (8 float4 per row)
            const int kq = q & 7;              // float4 index within 32 floats
            float4 av = ((const float4*)(vn + (size_t)(rowBase + r) * 128 + kc))[kq];
            float4 bv = ((const float4*)(tn + (size_t)(colBase + r) * 128 + kc))[kq];
            *((float4*)&As[r * LDSW + kq * 4]) = av;
            *((float4*)&Bs[r * LDSW + kq * 4]) = bv;
        }
        __syncthreads();

        #pragma unroll
        for (int k0 = 0; k0 < KC; k0 += 4) {
            v2f afrag[2], bfrag[4];
            #pragma unroll
            for (int i = 0; i < 2; ++i)
                afrag[i] = *((const v2f*)&As[(wr * 32 + i * 16 + mfrag) * LDSW + k0 + hi * 2]);
            #pragma unroll
            for (int j = 0; j < 4; ++j)
                bfrag[j] = *((const v2f*)&Bs[(wc * 64 + j * 16 + mfrag) * LDSW + k0 + hi * 2]);
            #pragma unroll
            for (int i = 0; i < 2; ++i)
                #pragma unroll
                for (int j = 0; j < 4; ++j)
                    acc[i][j] = __builtin_amdgcn_wmma_f32_16x16x4_f32(
                        /*neg_a=*/false, afrag[i],
                        /*neg_b=*/false, bfrag[j],
                        /*c_mod=*/(short)0, acc[i][j],
                        /*reuse_a=*/false, /*reuse_b=*/false);
        }
    }

    // Epilogue: C/D layout -> row = g + 8*hi, col = lane&15 within each tile.
    #pragma unroll
    for (int i = 0; i < 2; ++i) {
        #pragma unroll
        for (int j = 0; j < 4; ++j) {
            const int tileRow = rowBase + wr * 32 + i * 16 + 8 * hi;
            const int tileCol = colBase + wc * 64 + j * 16 + mfrag;
            #pragma unroll
            for (int g = 0; g < 8; ++g) {
                out[(size_t)(tileRow + g) * 8192 + tileCol] = acc[i][j][g];
            }
        }
    }
}

// ---------------------------------------------------------------------------
// Launcher
// ---------------------------------------------------------------------------
extern "C" void kernel_launch(void* const* d_in, const int* in_sizes, int n_in,
                              void* d_out, int out_size, void* d_ws, size_t ws_size,
                              hipStream_t stream) {
    const float* vision = (const float*)d_in[0];   // [8192, 128]
    const float* textp  = (const float*)d_in[1];   // [8192, 768]
    const float* projw  = (const float*)d_in[2];   // [128, 768]
    const float* projb  = (const float*)d_in[3];   // [128]
    float* out = (float*)d_out;                    // [8192, 8192]

    float* vn = (float*)d_ws;                      // [8192, 128]  4 MB
    float* tn = vn + (size_t)8192 * 128;           // [8192, 128]  4 MB

    proj_norm_kernel<<<8192, 128, 0, stream>>>(textp, projw, projb, tn);
    vision_norm_kernel<<<1024, 256, 0, stream>>>(vision, vn);

    dim3 grid(64, 64, 1);
    cosine_gemm_kernel<<<grid, dim3(256, 1, 1), 0, stream>>>(vn, tn, out);
}